// MambaBlock_83588653515375
// MI455X (gfx1250) — compile-verified
//
#include <hip/hip_runtime.h>
#include <hip/hip_bf16.h>
#include <stdint.h>

typedef __attribute__((ext_vector_type(2))) float v2f;
typedef __attribute__((ext_vector_type(8))) float v8f;

#define D_MODEL 768
#define D_INNER 1536
#define D_STATE 16
#define DT_RANK 48
#define SEQ     1024
#define NTOK    2048   // BATCH * SEQ
#define D_XZ    3072   // 2 * D_INNER
#define D_DBC   80     // DT_RANK + 2*D_STATE

// --- CDNA5 async global->LDS copy (ASYNCcnt path), with safe fallback -------
#if defined(__gfx1250__) && __has_builtin(__builtin_amdgcn_global_load_async_to_lds_b128)
#define HAVE_ASYNC 1
typedef __attribute__((__vector_size__(16))) int i4vec;
typedef __attribute__((address_space(1))) i4vec* gptr_v4i;   // global
typedef __attribute__((address_space(3))) i4vec* lptr_v4i;   // LDS
#else
#define HAVE_ASYNC 0
#endif

__device__ __forceinline__ void wait_async_lds() {
#if HAVE_ASYNC
#if __has_builtin(__builtin_amdgcn_s_wait_asynccnt)
    __builtin_amdgcn_s_wait_asynccnt(0);
#else
    asm volatile("s_wait_asynccnt 0x0" ::: "memory");
#endif
#endif
}

// ---------------------------------------------------------------------------
// RMSNorm: one block per token, 256 threads, LDS tree reduction.
// ---------------------------------------------------------------------------
__global__ void k_rmsnorm(const float* __restrict__ x,
                          const float* __restrict__ w,
                          float* __restrict__ out) {
    __shared__ float red[256];
    const int tok = blockIdx.x;
    const float* xr = x + (size_t)tok * D_MODEL;
    float s = 0.f;
    for (int i = threadIdx.x; i < D_MODEL; i += 256) {
        float v = xr[i];
        s += v * v;
    }
    red[threadIdx.x] = s;
    __syncthreads();
    for (int off = 128; off > 0; off >>= 1) {
        if (threadIdx.x < off) red[threadIdx.x] += red[threadIdx.x + off];
        __syncthreads();
    }
    const float inv = rsqrtf(red[0] / (float)D_MODEL + 1e-6f);
    float* orow = out + (size_t)tok * D_MODEL;
    for (int i = threadIdx.x; i < D_MODEL; i += 256) {
        orow[i] = xr[i] * inv * w[i];
    }
}

// ---------------------------------------------------------------------------
// fp32 WMMA GEMM, LDS-staged double-buffered B tiles, MSUB M-subtiles/wave.
// Block = 256 threads (8 waves) computes a (128*MSUB) x (16*NSUB) tile of C.
//   grid.x = N / (16*NSUB), grid.y = M / (128*MSUB).
// Each wave owns MSUB 16-row subtiles: one B fragment feeds MSUB WMMAs,
// halving LDS read pressure per FLOP vs MSUB=1. All strides compile-time.
// EPI: 0 = none, 1 = +bias then softplus, 2 = +residual (ld = LDC).
// ---------------------------------------------------------------------------
template <int MSUB, int NSUB, int KC, int K, int LDA, int LDB, int LDC, int EPI>
__global__ void __launch_bounds__(256)
k_gemm_wmma(const float* __restrict__ A, const float* __restrict__ B,
            float* __restrict__ C, const float* __restrict__ bias,
            const float* __restrict__ residual) {
    constexpr int NB  = 16 * NSUB;     // block tile width
    constexpr int NBP = NB + 4;        // padded LDS row (float4-aligned)
    constexpr int NCH = K / KC;        // K chunks
    static_assert(K % KC == 0 && KC % 4 == 0, "bad KC");

    __shared__ float sB[2][KC][NBP];

    const int tid  = threadIdx.x;
    const int lane = tid & 31;
    const int wv   = tid >> 5;
    const int half = lane >> 4;        // 0 | 1
    const int mr   = lane & 15;
    const int n0   = blockIdx.x * NB;
    const int m0   = blockIdx.y * (128 * MSUB) + wv * (16 * MSUB);

    v8f acc[MSUB][NSUB] = {};

    // cooperative stage of B chunk c into buffer buf (float4 granularity)
    auto stage = [&](int c, int buf) {
        constexpr int NV4 = KC * NB / 4;
        for (int idx = tid; idx < NV4; idx += 256) {
            const int r  = idx / (NB / 4);
            const int cg = idx % (NB / 4);
            const float* gp = B + (size_t)(c * KC + r) * LDB + n0 + cg * 4;
            float* lp = &sB[buf][r][cg * 4];
#if HAVE_ASYNC
            __builtin_amdgcn_global_load_async_to_lds_b128(
                (gptr_v4i)(uintptr_t)gp, (lptr_v4i)(uintptr_t)lp, 0, 0);
#else
            *(float4*)lp = *(const float4*)gp;
#endif
        }
    };

    // compute one K chunk from LDS buffer buf
    auto compute = [&](int c, int buf) {
        v2f af[MSUB][KC / 4];
#pragma unroll
        for (int m = 0; m < MSUB; ++m) {         // prefetch A fragments
            const float* Ab = A + (size_t)(m0 + m * 16 + mr) * LDA
                              + c * KC + 2 * half;
#pragma unroll
            for (int j = 0; j < KC / 4; ++j) {
                af[m][j].x = Ab[4 * j];
                af[m][j].y = Ab[4 * j + 1];
            }
        }
#pragma unroll
        for (int j = 0; j < KC / 4; ++j) {
            const float* Br = &sB[buf][4 * j + 2 * half][0];
#pragma unroll
            for (int s = 0; s < NSUB; ++s) {
                v2f bv;
                bv.x = Br[s * 16 + mr];          // K row 4j+2h
                bv.y = Br[NBP + s * 16 + mr];    // K row 4j+2h+1
#pragma unroll
                for (int m = 0; m < MSUB; ++m) {
                    acc[m][s] = __builtin_amdgcn_wmma_f32_16x16x4_f32(
                        false, af[m][j], false, bv, (short)0, acc[m][s],
                        false, false);
                }
            }
        }
    };

    stage(0, 0);
    wait_async_lds();
    __syncthreads();
#pragma unroll 1
    for (int c = 0; c < NCH; ++c) {
        if (c + 1 < NCH) stage(c + 1, (c + 1) & 1);  // overlap with compute
        compute(c, c & 1);
        wait_async_lds();
        __syncthreads();
    }

    // epilogue + store
#pragma unroll
    for (int m = 0; m < MSUB; ++m) {
#pragma unroll
        for (int s = 0; s < NSUB; ++s) {
            const int col = n0 + s * 16 + mr;
#pragma unroll
            for (int v = 0; v < 8; ++v) {
                const int row = m0 + m * 16 + v + 8 * half;
                float val = acc[m][s][v];
                if (EPI == 1) {
                    val += bias[col];
                    val = (val > 20.f) ? val : log1pf(expf(val));  // softplus
                } else if (EPI == 2) {
                    val += residual[(size_t)row * LDC + col];
                }
                C[(size_t)row * LDC + col] = val;
            }
        }
    }
}

// ---------------------------------------------------------------------------
// Causal depthwise conv (width 4) over xi half of xz, + bias, + SiLU.
// ---------------------------------------------------------------------------
__global__ void k_conv_silu(const float* __restrict__ xz,
                            const float* __restrict__ cw,
                            const float* __restrict__ cb,
                            float* __restrict__ xc) {
    const int idx = blockIdx.x * blockDim.x + threadIdx.x;
    if (idx >= 2 * SEQ * D_INNER) return;
    const int d = idx % D_INNER;
    const int t = (idx / D_INNER) % SEQ;
    const int b = idx / (D_INNER * SEQ);
    float acc = cb[d];
#pragma unroll
    for (int k = 0; k < 4; ++k) {
        const int tt = t - 3 + k;
        if (tt >= 0)
            acc += xz[(size_t)(b * SEQ + tt) * D_XZ + d] * cw[d * 4 + k];
    }
    xc[idx] = acc / (1.f + expf(-acc));    // silu
}

// ---------------------------------------------------------------------------
// Selective scan: one thread per (batch, channel); 16 states in registers.
// B/C rows (shared by all channels) staged through LDS in 32-step chunks.
// Fuses y = sum_n h_n*C_n + D*xi, then gate by silu(z).
// ---------------------------------------------------------------------------
__global__ void k_scan(const float* __restrict__ dt,
                       const float* __restrict__ xc,
                       const float* __restrict__ xz,
                       const float* __restrict__ dbc,
                       const float* __restrict__ A_log,
                       const float* __restrict__ Dp,
                       float* __restrict__ yact) {
    __shared__ float sB[32][D_STATE];
    __shared__ float sC[32][D_STATE];
    const int d = blockIdx.x * blockDim.x + threadIdx.x;   // 0..D_INNER-1
    const int b = blockIdx.y;

    float acoef[D_STATE], h[D_STATE];
    const float Dd = Dp[d];
#pragma unroll
    for (int n = 0; n < D_STATE; ++n) {
        acoef[n] = -expf(A_log[d * D_STATE + n]);          // A = -exp(A_log)
        h[n] = 0.f;
    }

    for (int t0 = 0; t0 < SEQ; t0 += 32) {
        for (int r = 0; r < 4; ++r) {
            const int li = r * 256 + threadIdx.x;
            const int i = li >> 5;
            const int j = li & 31;
            const float v = dbc[(size_t)(b * SEQ + t0 + i) * D_DBC + DT_RANK + j];
            if (j < D_STATE) sB[i][j] = v;
            else             sC[i][j - D_STATE] = v;
        }
        __syncthreads();

        for (int i = 0; i < 32; ++i) {
            const size_t row = (size_t)(b * SEQ + t0 + i);
            const float dtv = dt[row * D_INNER + d];
            const float xiv = xc[row * D_INNER + d];
            const float zv  = xz[row * D_XZ + D_INNER + d];
            const float dtxi = dtv * xiv;
            float y = 0.f;
#pragma unroll
            for (int n = 0; n < D_STATE; ++n) {
                h[n] = expf(dtv * acoef[n]) * h[n] + dtxi * sB[i][n];
                y += h[n] * sC[i][n];
            }
            y += Dd * xiv;
            y *= zv / (1.f + expf(-zv));   // * silu(z)
            yact[row * D_INNER + d] = y;
        }
        __syncthreads();
    }
}

// ---------------------------------------------------------------------------
extern "C" void kernel_launch(void* const* d_in, const int* in_sizes, int n_in,
                              void* d_out, int out_size, void* d_ws, size_t ws_size,
                              hipStream_t stream) {
    (void)in_sizes; (void)n_in; (void)out_size; (void)ws_size;

    const float* x        = (const float*)d_in[0];
    const float* norm_w   = (const float*)d_in[1];
    const float* in_proj  = (const float*)d_in[2];
    const float* conv_w   = (const float*)d_in[3];
    const float* conv_b   = (const float*)d_in[4];
    const float* x_proj   = (const float*)d_in[5];
    const float* dt_proj  = (const float*)d_in[6];
    const float* dt_b     = (const float*)d_in[7];
    const float* A_log    = (const float*)d_in[8];
    const float* Dp       = (const float*)d_in[9];
    const float* out_proj = (const float*)d_in[10];
    float* out = (float*)d_out;

    float* ws   = (float*)d_ws;
    float* h    = ws;                              // 2048*768
    float* xz   = h    + (size_t)NTOK * D_MODEL;   // 2048*3072
    float* xc   = xz   + (size_t)NTOK * D_XZ;      // 2048*1536
    float* dbc  = xc   + (size_t)NTOK * D_INNER;   // 2048*80
    float* dtb  = dbc  + (size_t)NTOK * D_DBC;     // 2048*1536
    float* yact = dtb  + (size_t)NTOK * D_INNER;   // 2048*1536

    // 1) RMSNorm
    k_rmsnorm<<<NTOK, 256, 0, stream>>>(x, norm_w, h);

    // 2) in_proj: [2048,768] @ [768,3072] -> xz   (256x64 block tiles)
    k_gemm_wmma<2, 4, 32, D_MODEL, D_MODEL, D_XZ, D_XZ, 0>
        <<<dim3(D_XZ / 64, NTOK / 256), 256, 0, stream>>>(
            h, in_proj, xz, nullptr, nullptr);

    // 3) causal depthwise conv + SiLU -> xc
    {
        const int n = 2 * SEQ * D_INNER;
        k_conv_silu<<<(n + 255) / 256, 256, 0, stream>>>(xz, conv_w, conv_b, xc);
    }

    // 4) x_proj: [2048,1536] @ [1536,80] -> dbc (NSUB=5 covers all 80 cols)
    k_gemm_wmma<2, 5, 32, D_INNER, D_INNER, D_DBC, D_DBC, 0>
        <<<dim3(1, NTOK / 256), 256, 0, stream>>>(
            xc, x_proj, dbc, nullptr, nullptr);

    // 5) dt_proj: dt_low [2048,48] @ [48,1536], +bias, softplus -> dtb
    k_gemm_wmma<2, 4, 16, DT_RANK, D_DBC, D_INNER, D_INNER, 1>
        <<<dim3(D_INNER / 64, NTOK / 256), 256, 0, stream>>>(
            dbc, dt_proj, dtb, dt_b, nullptr);

    // 6) selective scan + D*xi + silu(z) gate -> yact
    k_scan<<<dim3(D_INNER / 256, 2), 256, 0, stream>>>(
        dtb, xc, xz, dbc, A_log, Dp, yact);

    // 7) out_proj: [2048,1536] @ [1536,768] + residual -> out
    k_gemm_wmma<2, 4, 32, D_INNER, D_INNER, D_MODEL, D_MODEL, 2>
        <<<dim3(D_MODEL / 64, NTOK / 256), 256, 0, stream>>>(
            yact, out_proj, out, nullptr, x);
}